// MambaNet_74500502716714
// MI455X (gfx1250) — compile-verified
//
#include <hip/hip_runtime.h>
#include <math.h>

#define D_   128
#define ED_  256
#define N_   16
#define R_   8
#define KC_  4
#define NL_  3
#define B_   4
#define L_   1024
#define HID_ 256
#define OUT_ 2
#define MTOK (B_*L_)      // 4096 tokens
#define DBCW 40           // R + 2N
#define EPS_ 1e-5f

typedef __attribute__((ext_vector_type(2))) float v2f;
typedef __attribute__((ext_vector_type(8))) float v8f;

__device__ __forceinline__ float silu_(float x)     { return x / (1.f + __expf(-x)); }
__device__ __forceinline__ float softplus_(float x) { return x > 20.f ? x : log1pf(__expf(x)); }

// ---------------------------------------------------------------------------
// C[M x N] = A[M x KD] (row-major, leading dim lda) * W[N x KD]^T + bias
// One wave32 per 16x16 output tile; V_WMMA_F32_16X16X4_F32, K stepped by 4.
// EPI: 0 = plain(+bias), 1 = softplus, 2 = residual add into C, 3 = leaky relu
// BOUNDED: N is not a multiple of 16 -> mask B loads / stores (branch-free).
// KD: compile-time K for unrolling.
// ---------------------------------------------------------------------------
template<int EPI, bool BOUNDED, int KD>
__global__ void gemm_wmma_f32(const float* __restrict__ A, int lda,
                              const float* __restrict__ W,
                              const float* __restrict__ bias,
                              float* __restrict__ C,
                              int N)
{
  const int n0   = blockIdx.x * 16;
  const int m0   = blockIdx.y * 16;
  const int lane = threadIdx.x & 31;
  const int row  = lane & 15;
  const int hi   = lane >> 4;

  v8f acc = {};
  const float* Arow  = A + (size_t)(m0 + row) * lda + 2*hi;
  const bool  bvalid = !BOUNDED || (n0 + row) < N;
  // clamp to row 0 so the load is always in-bounds; zero via mask instead of branch
  const float* Wrow  = W + (size_t)(bvalid ? (n0 + row) : 0) * KD + 2*hi;
  const float msk    = bvalid ? 1.f : 0.f;

#pragma unroll 8
  for (int k = 0; k < KD; k += 4) {
    // A 16x4: VGPR0 holds K = k+2*hi, VGPR1 holds K = k+2*hi+1 (ISA 7.12.2)
    v2f a = *(const v2f*)(Arow + k);
    v2f b = *(const v2f*)(Wrow + k);
    if (BOUNDED) { b.x *= msk; b.y *= msk; }     // folds away when !BOUNDED
    acc = __builtin_amdgcn_wmma_f32_16x16x4_f32(false, a, false, b,
                                                (short)0, acc, false, false);
  }

  const int col = n0 + row;
  if (!BOUNDED || col < N) {
    const float bv = bias ? bias[col] : 0.f;
#pragma unroll
    for (int v = 0; v < 8; ++v) {
      const int r = m0 + v + 8*hi;          // C/D: VGPR v -> M = v (lo half) / v+8 (hi half)
      size_t off = (size_t)r * N + col;
      float val = acc[v] + bv;
      if      (EPI == 1) val = softplus_(val);
      else if (EPI == 2) val = C[off] + val;
      else if (EPI == 3) val = val > 0.f ? val : 0.01f * val;
      C[off] = val;
    }
  }
}

// cur0 = x ; cur1 = flip(x, axis=time)
__global__ void init_streams(const float* __restrict__ x,
                             float* __restrict__ cur0, float* __restrict__ cur1)
{
  int idx = blockIdx.x * 256 + threadIdx.x;            // over B*L*D
  int c = idx % D_;
  int t = (idx / D_) % L_;
  int b = idx / (D_ * L_);
  cur0[idx] = x[idx];
  cur1[idx] = x[((size_t)b * L_ + (L_ - 1 - t)) * D_ + c];
}

// RMS norm: one wave per token (D=128 -> 4 elems/lane), shfl_xor reduction
__global__ void rms_kernel(const float* __restrict__ x, const float* __restrict__ w,
                           float* __restrict__ out)
{
  int token = blockIdx.x * 8 + (threadIdx.x >> 5);
  int lane  = threadIdx.x & 31;
  const float* xr = x + (size_t)token * D_;
  float v0 = xr[lane], v1 = xr[lane+32], v2 = xr[lane+64], v3 = xr[lane+96];
  float s = v0*v0 + v1*v1 + v2*v2 + v3*v3;
#pragma unroll
  for (int o = 16; o > 0; o >>= 1) s += __shfl_xor(s, o, 32);
  float r = rsqrtf(s * (1.f / D_) + EPS_);
  float* orow = out + (size_t)token * D_;
  orow[lane]    = v0 * r * w[lane];
  orow[lane+32] = v1 * r * w[lane+32];
  orow[lane+64] = v2 * r * w[lane+64];
  orow[lane+96] = v3 * r * w[lane+96];
}

// depthwise causal conv (K=4) over xz[..., :ED] + bias, then SiLU
__global__ void conv_silu(const float* __restrict__ xz, const float* __restrict__ cw,
                          const float* __restrict__ cb, float* __restrict__ xic)
{
  int idx = blockIdx.x * 256 + threadIdx.x;            // over MTOK*ED
  int e   = idx % ED_;
  int tok = idx / ED_;
  int t   = tok % L_;
  float acc = cb[e];
#pragma unroll
  for (int k = 0; k < KC_; ++k) {
    int tt = t - (KC_ - 1) + k;
    if (tt >= 0)
      acc += xz[(size_t)(tok - (KC_ - 1) + k) * (2*ED_) + e] * cw[e*KC_ + k];
  }
  xic[idx] = silu_(acc);
}

// selective-scan: one block per batch, one thread per channel e, 16 states in regs.
// B_t / C_t (32 floats) are wave-broadcast: each lane loads one value, the
// unrolled n-loop pulls them with __shfl -> zero LDS, zero barriers.
__global__ void ssm_kernel(const float* __restrict__ delta, const float* __restrict__ xic,
                           const float* __restrict__ dbc,  const float* __restrict__ A_log,
                           const float* __restrict__ Dp,   float* __restrict__ y)
{
  int b    = blockIdx.x;
  int e    = threadIdx.x;
  int lane = e & 31;
  float Ae[N_], h[N_];
#pragma unroll
  for (int n = 0; n < N_; ++n) { Ae[n] = -__expf(A_log[e*N_ + n]); h[n] = 0.f; }
  const float dpe = Dp[e];
  for (int t = 0; t < L_; ++t) {
    int tok = b * L_ + t;
    float bcv = dbc[(size_t)tok * DBCW + R_ + lane];   // lane n: B_t[n]; lane 16+n: C_t[n]
    float de  = delta[(size_t)tok * ED_ + e];
    float xe  = xic  [(size_t)tok * ED_ + e];
    float dx  = de * xe;
    float acc = 0.f;
#pragma unroll
    for (int n = 0; n < N_; ++n) {
      float Bn = __shfl(bcv, n, 32);
      float Cn = __shfl(bcv, N_ + n, 32);
      h[n] = __expf(de * Ae[n]) * h[n] + dx * Bn;
      acc += h[n] * Cn;
    }
    y[(size_t)tok * ED_ + e] = acc + dpe * xe;
  }
}

// y *= silu(z) with z = xz[..., ED:]
__global__ void gate_silu(float* __restrict__ y, const float* __restrict__ xz)
{
  int idx = blockIdx.x * 256 + threadIdx.x;
  int e   = idx % ED_;
  int tok = idx / ED_;
  y[idx] = y[idx] * silu_(xz[(size_t)tok * (2*ED_) + ED_ + e]);
}

// h = concat(fwd, flip(bwd))
__global__ void concat_head(const float* __restrict__ cur0, const float* __restrict__ cur1,
                            float* __restrict__ h)
{
  int idx = blockIdx.x * 256 + threadIdx.x;            // over MTOK * 2D
  int c   = idx % (2*D_);
  int tok = idx / (2*D_);
  int b = tok / L_, t = tok % L_;
  float v;
  if (c < D_) v = cur0[(size_t)tok * D_ + c];
  else        v = cur1[((size_t)b * L_ + (L_ - 1 - t)) * D_ + (c - D_)];
  h[idx] = v;
}

// per-column batchnorm stats -> fused scale/shift
__global__ void bn_stats(const float* __restrict__ h, int M, int C,
                         const float* __restrict__ g, const float* __restrict__ b,
                         float* __restrict__ scale, float* __restrict__ shift)
{
  int j = blockIdx.x, tid = threadIdx.x;
  float s = 0.f, s2 = 0.f;
  for (int r = tid; r < M; r += blockDim.x) {
    float v = h[(size_t)r * C + j];
    s += v; s2 += v * v;
  }
  __shared__ float ss[256], ss2[256];
  ss[tid] = s; ss2[tid] = s2; __syncthreads();
  for (int o = 128; o > 0; o >>= 1) {
    if (tid < o) { ss[tid] += ss[tid+o]; ss2[tid] += ss2[tid+o]; }
    __syncthreads();
  }
  if (tid == 0) {
    float m   = ss[0] / M;
    float var = ss2[0] / M - m * m;
    float inv = rsqrtf(var + EPS_);
    scale[j] = g[j] * inv;
    shift[j] = b[j] - m * g[j] * inv;
  }
}

__global__ void bn_apply(const float* __restrict__ h, const float* __restrict__ scale,
                         const float* __restrict__ shift, float* __restrict__ out, int C)
{
  int idx = blockIdx.x * 256 + threadIdx.x;
  int j = idx % C;
  out[idx] = h[idx] * scale[j] + shift[j];
}

// ---------------------------------------------------------------------------
extern "C" void kernel_launch(void* const* d_in, const int* in_sizes, int n_in,
                              void* d_out, int out_size, void* d_ws, size_t ws_size,
                              hipStream_t stream)
{
  (void)in_sizes; (void)n_in; (void)out_size; (void)ws_size;
  const float* x       = (const float*)d_in[0];
  const float* rms_w   = (const float*)d_in[1];
  const float* in_w    = (const float*)d_in[2];
  const float* in_b    = (const float*)d_in[3];
  const float* conv_w  = (const float*)d_in[4];
  const float* conv_b  = (const float*)d_in[5];
  const float* xproj_w = (const float*)d_in[6];
  const float* dt_w    = (const float*)d_in[7];
  const float* dt_b    = (const float*)d_in[8];
  const float* A_log   = (const float*)d_in[9];
  const float* Dp      = (const float*)d_in[10];
  const float* out_w   = (const float*)d_in[11];
  const float* out_b   = (const float*)d_in[12];
  const float* bn1_g = (const float*)d_in[13];
  const float* bn1_b = (const float*)d_in[14];
  const float* w1    = (const float*)d_in[15];
  const float* b1    = (const float*)d_in[16];
  const float* bn2_g = (const float*)d_in[17];
  const float* bn2_b = (const float*)d_in[18];
  const float* w2    = (const float*)d_in[19];
  const float* b2    = (const float*)d_in[20];
  const float* bn3_g = (const float*)d_in[21];
  const float* bn3_b = (const float*)d_in[22];
  const float* w3    = (const float*)d_in[23];
  const float* b3    = (const float*)d_in[24];

  float* ws = (float*)d_ws;
  float* cur0  = ws;                         // 524288
  float* cur1  = ws + 524288;                // 524288
  float* xn    = ws + 1048576;               // 524288
  float* xz    = ws + 1572864;               // 2097152
  float* xic   = ws + 3670016;               // 1048576
  float* dbc   = ws + 4718592;               // 4096*40 = 163840
  float* delta = ws + 4882432;               // 1048576
  float* yb    = ws + 5931008;               // 1048576
  // head reuses freed regions:
  float* hh    = xz;                         // 1048576
  float* hn    = xz + 1048576;               // 1048576
  float* h2    = xic;                        // 1048576
  float* scale = dbc;                        // 256
  float* shift = dbc + 256;                  // 256

  init_streams<<<(MTOK * D_) / 256, 256, 0, stream>>>(x, cur0, cur1);

  for (int d = 0; d < 2; ++d) {
    float* cur = (d == 0) ? cur0 : cur1;
    for (int l = 0; l < NL_; ++l) {
      const int dl = d * NL_ + l;
      const float* rw = rms_w   + (size_t)dl * D_;
      const float* iw = in_w    + (size_t)dl * 2 * ED_ * D_;
      const float* ib = in_b    + (size_t)dl * 2 * ED_;
      const float* cw = conv_w  + (size_t)dl * ED_ * KC_;
      const float* cb = conv_b  + (size_t)dl * ED_;
      const float* xw = xproj_w + (size_t)dl * DBCW * ED_;
      const float* dw = dt_w    + (size_t)dl * ED_ * R_;
      const float* db = dt_b    + (size_t)dl * ED_;
      const float* al = A_log   + (size_t)dl * ED_ * N_;
      const float* dp = Dp      + (size_t)dl * ED_;
      const float* ow = out_w   + (size_t)dl * D_ * ED_;
      const float* ob = out_b   + (size_t)dl * D_;

      rms_kernel<<<MTOK / 8, 256, 0, stream>>>(cur, rw, xn);
      // xz = xn @ in_w^T + in_b              (4096 x 512, K=128)
      gemm_wmma_f32<0, false, D_><<<dim3((2*ED_)/16, MTOK/16), 32, 0, stream>>>(
          xn, D_, iw, ib, xz, 2*ED_);
      conv_silu<<<(MTOK * ED_) / 256, 256, 0, stream>>>(xz, cw, cb, xic);
      // dbc = xic @ xproj_w^T                (4096 x 40, K=256, no bias)
      gemm_wmma_f32<0, true, ED_><<<dim3((DBCW + 15)/16, MTOK/16), 32, 0, stream>>>(
          xic, ED_, xw, nullptr, dbc, DBCW);
      // delta = softplus(dbc[:, :8] @ dt_w^T + dt_b)  (4096 x 256, K=8)
      gemm_wmma_f32<1, false, R_><<<dim3(ED_/16, MTOK/16), 32, 0, stream>>>(
          dbc, DBCW, dw, db, delta, ED_);
      ssm_kernel<<<B_, ED_, 0, stream>>>(delta, xic, dbc, al, dp, yb);
      gate_silu<<<(MTOK * ED_) / 256, 256, 0, stream>>>(yb, xz);
      // cur += yb @ out_w^T + out_b          (4096 x 128, K=256, residual)
      gemm_wmma_f32<2, false, ED_><<<dim3(D_/16, MTOK/16), 32, 0, stream>>>(
          yb, ED_, ow, ob, cur, D_);
    }
  }

  // ---- head ----
  concat_head<<<(MTOK * 2 * D_) / 256, 256, 0, stream>>>(cur0, cur1, hh);

  bn_stats<<<2*D_, 256, 0, stream>>>(hh, MTOK, 2*D_, bn1_g, bn1_b, scale, shift);
  bn_apply<<<(MTOK * 2*D_) / 256, 256, 0, stream>>>(hh, scale, shift, hn, 2*D_);
  gemm_wmma_f32<3, false, 2*D_><<<dim3(HID_/16, MTOK/16), 32, 0, stream>>>(
      hn, 2*D_, w1, b1, h2, HID_);

  bn_stats<<<HID_, 256, 0, stream>>>(h2, MTOK, HID_, bn2_g, bn2_b, scale, shift);
  bn_apply<<<(MTOK * HID_) / 256, 256, 0, stream>>>(h2, scale, shift, hn, HID_);
  gemm_wmma_f32<3, false, HID_><<<dim3(HID_/16, MTOK/16), 32, 0, stream>>>(
      hn, HID_, w2, b2, hh, HID_);

  bn_stats<<<HID_, 256, 0, stream>>>(hh, MTOK, HID_, bn3_g, bn3_b, scale, shift);
  bn_apply<<<(MTOK * HID_) / 256, 256, 0, stream>>>(hh, scale, shift, hn, HID_);
  gemm_wmma_f32<0, true, HID_><<<dim3(1, MTOK/16), 32, 0, stream>>>(
      hn, HID_, w3, b3, (float*)d_out, OUT_);
}